// TSF_44461501448859
// MI455X (gfx1250) — compile-verified
//
#include <hip/hip_runtime.h>

// Problem constants (from the reference): B=8, C=512, T=64, W=14, H=14, N=3
#define B_   8
#define C_   512
#define T_   64
#define W_   14
#define H_   14
#define N_   3
#define K_   12544          // T*W*H, the contraction length
#define EPSF 1e-6f

#define BK    256           // K-tile staged in LDS per iteration (K_ = 49*BK)
#define LDSS  260           // padded LDS row stride (floats): 260 % 64 banks = 4
                            // -> 16-row strided b64 reads hit all 64 banks once

typedef __attribute__((ext_vector_type(2))) float v2f;   // A/B frag: 16x4 / 4x16 f32
typedef __attribute__((ext_vector_type(8))) float v8f;   // C/D frag: 16x16 f32

// ---------------------------------------------------------------------------
// Kernel 1: normalized Gaussian factors F[b][n][k], n padded to 4 (row 3 = 0)
// ---------------------------------------------------------------------------
__global__ __launch_bounds__(256) void tsf_factor_kernel(
    const float* __restrict__ length,
    const float* __restrict__ mu_t,  const float* __restrict__ mu_x,
    const float* __restrict__ mu_y,  const float* __restrict__ sigma_t,
    const float* __restrict__ sigma_x, const float* __restrict__ sigma_y,
    float* __restrict__ F)
{
    const int n   = blockIdx.x;           // 0..3 (3 == zero-pad row)
    const int b   = blockIdx.y;           // 0..7
    const int tid = threadIdx.x;
    float* Frow = F + ((size_t)b * 4 + n) * K_;

    if (n == 3) {                          // zero padding row for the B matrix
        for (int idx = tid; idx < K_; idx += 256) Frow[idx] = 0.0f;
        return;
    }

    // scalar parameter transforms (all threads compute the same cheap values)
    const float st = expf(1.5f - 2.0f * fabsf(tanhf(sigma_t[n])));
    const float sx = expf(1.5f - 2.0f * fabsf(tanhf(sigma_x[n])));
    const float sy = expf(1.5f - 2.0f * fabsf(tanhf(sigma_y[n])));
    const float pt = 1.0f / (st * st + EPSF);
    const float px = 1.0f / (sx * sx + EPSF);
    const float py = 1.0f / (sy * sy + EPSF);
    const float mt = (length[b] - 1.0f) * (tanhf(mu_t[n]) + 1.0f) * 0.5f;
    const float mx = (float)(W_ - 1)    * (tanhf(mu_x[n]) + 1.0f) * 0.5f;
    const float my = (float)(H_ - 1)    * (tanhf(mu_y[n]) + 1.0f) * 0.5f;

    float lsum = 0.0f;
    for (int idx = tid; idx < K_; idx += 256) {      // K_/256 = 49 iterations
        const int t   = idx / (W_ * H_);
        const int rem = idx - t * (W_ * H_);
        const int x   = rem / H_;
        const int y   = rem - x * H_;
        const float dt = (float)t - mt;
        const float dx = (float)x - mx;
        const float dy = (float)y - my;
        const float e  = expf(-0.5f * (pt * dt * dt + px * dx * dx + py * dy * dy));
        Frow[idx] = e;
        lsum += e;
    }

    __shared__ float rbuf[256];
    rbuf[tid] = lsum;
    __syncthreads();
    #pragma unroll
    for (int s = 128; s > 0; s >>= 1) {
        if (tid < s) rbuf[tid] += rbuf[tid + s];
        __syncthreads();
    }
    const float scale = 1.0f / (rbuf[0] + EPSF);
    for (int idx = tid; idx < K_; idx += 256) Frow[idx] *= scale;
}

// ---------------------------------------------------------------------------
// Kernel 2: O[b, c0:c0+16, 0:3] = sum_k V[b, c, k] * F[b, n, k]
// via v_wmma_f32_16x16x4_f32, K streamed through LDS in 256-wide tiles.
// Grid: (C/16, B); 256 threads = 8 waves; wave w owns k-slice [w*32, w*32+32).
// Both A and B fragments come from LDS -> the WMMA loop has no branches and
// no exec-mask manipulation (keeps the wave at full EXEC as WMMA requires).
// ---------------------------------------------------------------------------
__global__ __launch_bounds__(256) void tsf_pool_wmma_kernel(
    const float* __restrict__ video,   // (B, C, K) f32
    const float* __restrict__ F,       // (B, 4, K) f32, row 3 zero
    float* __restrict__ out)           // (B, C, N) f32
{
    __shared__ float sV[16 * LDSS];    // 16 c-rows x BK k-cols, padded stride
    __shared__ float sF[16 * LDSS];    // 16 n-rows (rows 4..15 stay zero)
    __shared__ float sRed[8 * 256];    // per-wave 16x16 partial tiles

    const int tid  = threadIdx.x;
    const int wave = tid >> 5;
    const int lane = tid & 31;
    const int l16  = lane & 15;        // A: row M / B: col N
    const int hv   = lane >> 4;        // half selects K-pair {0,1} vs {2,3}
    const int c0   = blockIdx.x * 16;
    const int b    = blockIdx.y;

    const float* Vt = video + ((size_t)b * C_ + c0) * K_;
    const float* Fb = F + (size_t)b * 4 * K_;

    // coalesced video tile mapping: threads t..t+15 cover 256B contiguous
    const int lr = tid >> 4;                 // c-row 0..15
    const int lc = (tid & 15) << 2;          // col base (floats)
    const float* vrow = Vt + (size_t)lr * K_;

    // coalesced F tile mapping: 4 rows x 256 cols, one float4 per thread
    const int fr = tid >> 6;                 // n-row 0..3
    const int fc = (tid & 63) << 2;          // col base (floats)
    const float* frow = Fb + (size_t)fr * K_;

    // zero the padded B-matrix region once (rows 4..15 remain zero)
    for (int idx = tid; idx < 16 * LDSS; idx += 256) sF[idx] = 0.0f;

    v8f acc = {0.f, 0.f, 0.f, 0.f, 0.f, 0.f, 0.f, 0.f};

    for (int k0 = 0; k0 < K_; k0 += BK) {    // 49 tiles
        __syncthreads();                     // LDS reuse guard (covers init too)
        #pragma unroll
        for (int j = 0; j < 4; ++j) {        // video: 16 rows x 256 floats
            const int col = lc + j * 64;
            const float4 p = *(const float4*)(vrow + k0 + col);
            *(float4*)(&sV[lr * LDSS + col]) = p;
        }
        {                                    // F: 4 rows x 256 floats
            const float4 q = *(const float4*)(frow + k0 + fc);
            *(float4*)(&sF[fr * LDSS + fc]) = q;
        }
        if (k0 + BK < K_)                    // global_prefetch_b8 for next tile
            __builtin_prefetch(vrow + k0 + BK + lc, 0, 1);
        __syncthreads();

        const int kbase = wave << 5;         // this wave's 32-k slice
        #pragma unroll
        for (int s = 0; s < 8; ++s) {
            const int kk = kbase + (s << 2) + (hv << 1);
            // A 16x4: lane holds V[c0+l16][kk], V[c0+l16][kk+1]
            const v2f a  = *(const v2f*)(&sV[l16 * LDSS + kk]);
            // B 4x16: lane holds F[n=l16][kk], F[n=l16][kk+1]; rows >=4 are 0
            const v2f bf = *(const v2f*)(&sF[l16 * LDSS + kk]);
            acc = __builtin_amdgcn_wmma_f32_16x16x4_f32(
                false, a, false, bf, (short)0, acc, false, false);
        }
    }

    // C/D layout: acc[i] = element (M = i + hv*8, N = l16)
    #pragma unroll
    for (int i = 0; i < 8; ++i)
        sRed[wave * 256 + l16 * 16 + (i + hv * 8)] = acc[i];
    __syncthreads();

    const int M = tid & 15;                  // c within tile
    const int n = tid >> 4;                  // 0..15, keep n < 3
    float ssum = 0.0f;
    #pragma unroll
    for (int w = 0; w < 8; ++w) ssum += sRed[w * 256 + n * 16 + M];
    if (n < N_)
        out[((size_t)b * C_ + (c0 + M)) * N_ + n] = ssum;
}

// ---------------------------------------------------------------------------
extern "C" void kernel_launch(void* const* d_in, const int* in_sizes, int n_in,
                              void* d_out, int out_size, void* d_ws, size_t ws_size,
                              hipStream_t stream) {
    const float* video   = (const float*)d_in[0];
    const float* length  = (const float*)d_in[1];
    const float* mu_t    = (const float*)d_in[2];
    const float* mu_x    = (const float*)d_in[3];
    const float* mu_y    = (const float*)d_in[4];
    const float* sigma_t = (const float*)d_in[5];
    const float* sigma_x = (const float*)d_in[6];
    const float* sigma_y = (const float*)d_in[7];

    float* Fw  = (float*)d_ws;   // (B,4,K) f32 = 1.6 MB of workspace
    float* out = (float*)d_out;  // (B, C*N) f32, fully overwritten each call

    tsf_factor_kernel<<<dim3(4, B_), 256, 0, stream>>>(
        length, mu_t, mu_x, mu_y, sigma_t, sigma_x, sigma_y, Fw);
    tsf_pool_wmma_kernel<<<dim3(C_ / 16, B_), 256, 0, stream>>>(video, Fw, out);
}